// MambaExpert_23957327577734
// MI455X (gfx1250) — compile-verified
//
#include <hip/hip_runtime.h>
#include <hip/hip_bf16.h>
#include <math.h>

typedef float v2f __attribute__((ext_vector_type(2)));
typedef float v8f __attribute__((ext_vector_type(8)));

#define B_  32
#define S_  4096
#define D_  128
#define N_  32
#define CH_ 32              // s-chunks per batch for pooling
#define SCHUNK (S_ / CH_)   // 128 timesteps per chunk
#define LN_EPS 1e-5f

// D = A(16x4 f32) * B(4x16 f32) + C(16x16 f32), full-precision matrix pipe op.
__device__ __forceinline__ v8f wmma4(v2f a, v2f b, v8f c) {
    return __builtin_amdgcn_wmma_f32_16x16x4_f32(
        /*neg_a=*/false, a, /*neg_b=*/false, b,
        /*c_mod=*/(short)0, c, /*reuse_a=*/false, /*reuse_b=*/false);
}

// Raw LDS byte offset of a __shared__ object (addrspace(3) pointers are the
// 32-bit LDS offset on gfx1250).
__device__ __forceinline__ unsigned lds_off(const void* p) {
    return (unsigned)(__UINTPTR_TYPE__)(__attribute__((address_space(3))) const void*)p;
}

// 16-byte async copy global -> LDS, tracked with ASYNCcnt (CDNA5 path that
// bypasses VGPRs entirely).
__device__ __forceinline__ void async_copy_b128(unsigned lds_byte, const void* g) {
    asm volatile("global_load_async_to_lds_b128 %0, %1, off"
                 :: "v"(lds_byte),
                    "v"((unsigned long long)(__UINTPTR_TYPE__)g)
                 : "memory");
}

__device__ __forceinline__ void wait_asynccnt0() {
    asm volatile("s_wait_asynccnt 0x0" ::: "memory");
}

// Hardware tanh (CDNA5 transcendental). v_nop guards the documented TRANS-op
// co-execution hazard before the result is consumed.
__device__ __forceinline__ float hw_tanh(float x) {
    float y;
    asm volatile("v_tanh_f32 %0, %1\n\tv_nop" : "=v"(y) : "v"(x));
    return y;
}

// ---------------------------------------------------------------------------
// K1: pre[t,n] = sum_d e[t,d]*(Ws[n,d]+Wi[n,d]) + bs[n]+bi[n]
// 4 waves/WG, each wave owns a 16-token tile (2 WMMA n-tiles, K=128 -> 32 steps)
// ---------------------------------------------------------------------------
__global__ __launch_bounds__(128)
void k1_pre(const int* __restrict__ x, const float* __restrict__ emb,
            const float* __restrict__ Ws, const float* __restrict__ bs,
            const float* __restrict__ Wi, const float* __restrict__ bi,
            float* __restrict__ pre) {
    __shared__ __align__(16) float etile[4][16][D_];
    const int wid  = threadIdx.x >> 5;
    const int lane = threadIdx.x & 31;
    const int g    = lane >> 4;
    const int col  = lane & 15;
    const int tokbase = (blockIdx.x * 4 + wid) * 16;

    // gather 16 embedding rows (each lane: 4 contiguous floats per row)
    for (int r = 0; r < 16; ++r) {
        const int vi = x[tokbase + r];
        *(float4*)(&etile[wid][r][lane * 4]) =
            *(const float4*)(emb + (size_t)vi * D_ + lane * 4);
    }
    __syncthreads();

    const float bsum0 = bs[col]      + bi[col];
    const float bsum1 = bs[col + 16] + bi[col + 16];

    v8f c0 = {0.f,0.f,0.f,0.f,0.f,0.f,0.f,0.f};
    v8f c1 = c0;
    const int n0 = col, n1 = col + 16;
#pragma unroll
    for (int kk = 0; kk < 32; ++kk) {
        const int k = kk * 4 + 2 * g;
        v2f a  = *(const v2f*)(&etile[wid][col][k]);
        v2f b0 = *(const v2f*)(Ws + n0 * D_ + k);
        v2f w0 = *(const v2f*)(Wi + n0 * D_ + k);
        v2f b1 = *(const v2f*)(Ws + n1 * D_ + k);
        v2f w1 = *(const v2f*)(Wi + n1 * D_ + k);
        b0 += w0;
        b1 += w1;
        c0 = wmma4(a, b0, c0);
        c1 = wmma4(a, b1, c1);
    }
#pragma unroll
    for (int v = 0; v < 8; ++v) {
        const int t = tokbase + v + 8 * g;            // C/D row = vgpr + 8*half
        pre[(size_t)t * N_ + n0] = c0[v] + bsum0;
        pre[(size_t)t * N_ + n1] = c1[v] + bsum1;
    }
}

// ---------------------------------------------------------------------------
// K2: serial scan, one wave per batch. Lane n owns h[n]; full h replicated in
// registers via LDS broadcast. Inner product in v2f form -> v_pk_fma_f32;
// nonlinearity via hardware v_tanh_f32. This is the latency-critical kernel.
// ---------------------------------------------------------------------------
__global__ __launch_bounds__(32)
void k2_scan(const float* __restrict__ pre, const float* __restrict__ A,
             float* __restrict__ hs) {
    const int b = blockIdx.x;
    const int n = threadIdx.x;

    v2f Ar2[N_ / 2];
#pragma unroll
    for (int j = 0; j < N_ / 2; ++j)
        Ar2[j] = *(const v2f*)(A + n * N_ + 2 * j);

    __shared__ float hsh[N_];
    v2f hloc2[N_ / 2];
#pragma unroll
    for (int j = 0; j < N_ / 2; ++j) hloc2[j] = (v2f){0.f, 0.f};

    const float* pb = pre + (size_t)b * S_ * N_;
    float*       hb = hs  + (size_t)b * S_ * N_;

    for (int s = 0; s < S_; ++s) {
        const float p = pb[(size_t)s * N_ + n];
        v2f a0 = {p, 0.f}, a1 = {0.f, 0.f}, a2 = a1, a3 = a1;
#pragma unroll
        for (int j = 0; j < N_ / 2; j += 4) {   // 4 chains of v_pk_fma_f32
            a0 += Ar2[j]     * hloc2[j];
            a1 += Ar2[j + 1] * hloc2[j + 1];
            a2 += Ar2[j + 2] * hloc2[j + 2];
            a3 += Ar2[j + 3] * hloc2[j + 3];
        }
        const v2f as = (a0 + a1) + (a2 + a3);
        const float h = hw_tanh(as.x + as.y);
        hb[(size_t)s * N_ + n] = h;
        __syncthreads();                        // single-wave WG: ~free
        hsh[n] = h;
        __syncthreads();
#pragma unroll
        for (int j = 0; j < N_ / 2; ++j)
            hloc2[j] = *(const v2f*)(&hsh[2 * j]);
    }
}

// ---------------------------------------------------------------------------
// K3: out = hs*Wo^T + bo ; gate = sigmoid(e*Wg^T + bg) ; accumulate
// sum_s(out*gate) per (b, chunk). 8 waves/WG = 8 d-tiles over one token tile.
// Tiles staged with GLOBAL_LOAD_ASYNC_TO_LDS_B128 (ASYNCcnt-tracked DMA).
// ---------------------------------------------------------------------------
__global__ __launch_bounds__(256)
void k3_pool(const int* __restrict__ x, const float* __restrict__ emb,
             const float* __restrict__ hs,
             const float* __restrict__ Wo, const float* __restrict__ bo,
             const float* __restrict__ Wg, const float* __restrict__ bg,
             float* __restrict__ partial) {
    __shared__ __align__(16) float etile[16][D_];
    __shared__ __align__(16) float htile[16][N_];

    const int b    = blockIdx.x / CH_;
    const int ch   = blockIdx.x % CH_;
    const int wid  = threadIdx.x >> 5;
    const int lane = threadIdx.x & 31;
    const int g    = lane >> 4;
    const int col  = lane & 15;
    const int dc   = wid * 16 + col;             // this lane's output column d

    const float bo_c = bo[dc];
    const float bg_c = bg[dc];

    float acc = 0.f;
    const int sbase = ch * SCHUNK;

    for (int tt = 0; tt < SCHUNK / 16; ++tt) {
        const int tok0 = b * S_ + sbase + tt * 16;
        __syncthreads();
        {   // async stage: 16x128 e tile (32B/thread) + 16x32 hs tile
            const int r = threadIdx.x >> 4;        // 0..15
            const int j = (threadIdx.x & 15) * 8;  // 0..120
            const int vi = x[tok0 + r];
            const float* ge = emb + (size_t)vi * D_ + j;
            const unsigned le = lds_off(&etile[r][j]);
            async_copy_b128(le,      ge);
            async_copy_b128(le + 16, ge + 4);
            if (threadIdx.x < 128) {               // 512 floats, 16B/thread
                const float* gh = hs + (size_t)tok0 * N_ + threadIdx.x * 4;
                async_copy_b128(lds_off(&htile[0][0]) + threadIdx.x * 16, gh);
            }
        }
        wait_asynccnt0();
        __syncthreads();

        v8f gc = {0.f,0.f,0.f,0.f,0.f,0.f,0.f,0.f};
        v8f oc = gc;
#pragma unroll
        for (int kk = 0; kk < 32; ++kk) {          // gate GEMM, K=128
            const int k = kk * 4 + 2 * g;
            v2f a = *(const v2f*)(&etile[col][k]);
            v2f w = *(const v2f*)(Wg + (size_t)dc * D_ + k);  // Wg L2-resident
            gc = wmma4(a, w, gc);
        }
#pragma unroll
        for (int kk = 0; kk < 8; ++kk) {           // out GEMM, K=32
            const int k = kk * 4 + 2 * g;
            v2f a = *(const v2f*)(&htile[col][k]);
            v2f w = *(const v2f*)(Wo + (size_t)dc * N_ + k);
            oc = wmma4(a, w, oc);
        }
#pragma unroll
        for (int v = 0; v < 8; ++v) {              // rows m = v + 8g of tile
            const float o  = oc[v] + bo_c;
            const float gt = 1.f / (1.f + __expf(-(gc[v] + bg_c)));
            acc += o * gt;                          // partial over 8 rows
        }
    }
    acc += __shfl_xor(acc, 16);                     // combine row-halves per col
    if (lane < 16)
        partial[(size_t)ch * B_ * D_ + b * D_ + dc] = acc;
}

// ---------------------------------------------------------------------------
// K4: fixed-order chunk reduction, /S, LayerNorm over D=128
// ---------------------------------------------------------------------------
__global__ __launch_bounds__(128)
void k4_final(const float* __restrict__ partial, const float* __restrict__ gamma,
              const float* __restrict__ beta, float* __restrict__ out) {
    const int b = blockIdx.x, d = threadIdx.x;
    float s = 0.f;
    for (int c = 0; c < CH_; ++c)
        s += partial[(size_t)c * B_ * D_ + b * D_ + d];
    const float pooled = s / (float)S_;

    __shared__ float red[D_];
    red[d] = pooled;
    __syncthreads();
    for (int off = 64; off > 0; off >>= 1) {
        if (d < off) red[d] += red[d + off];
        __syncthreads();
    }
    const float mu = red[0] / (float)D_;
    __syncthreads();
    const float diff = pooled - mu;
    red[d] = diff * diff;
    __syncthreads();
    for (int off = 64; off > 0; off >>= 1) {
        if (d < off) red[d] += red[d + off];
        __syncthreads();
    }
    const float var = red[0] / (float)D_;
    out[b * D_ + d] = diff * rsqrtf(var + LN_EPS) * gamma[d] + beta[d];
}

// ---------------------------------------------------------------------------
extern "C" void kernel_launch(void* const* d_in, const int* in_sizes, int n_in,
                              void* d_out, int out_size, void* d_ws, size_t ws_size,
                              hipStream_t stream) {
    const int*   x     = (const int*)  d_in[0];
    const float* emb   = (const float*)d_in[1];
    const float* Ws    = (const float*)d_in[2];
    const float* bs    = (const float*)d_in[3];
    const float* Wi    = (const float*)d_in[4];
    const float* bi    = (const float*)d_in[5];
    const float* Wo    = (const float*)d_in[6];
    const float* bo    = (const float*)d_in[7];
    const float* Wg    = (const float*)d_in[8];
    const float* bg    = (const float*)d_in[9];
    const float* A     = (const float*)d_in[10];
    const float* gamma = (const float*)d_in[11];
    const float* beta  = (const float*)d_in[12];
    float* out = (float*)d_out;

    char*  ws  = (char*)d_ws;
    float* pre = (float*)ws;                               // 16 MB
    float* hs  = (float*)(ws + (size_t)B_ * S_ * N_ * 4);  // 16 MB
    float* partial = (float*)ws;   // reuses pre region (dead after k2), 512 KB

    k1_pre  <<<B_ * S_ / 64, 128, 0, stream>>>(x, emb, Ws, bs, Wi, bi, pre);
    k2_scan <<<B_,            32, 0, stream>>>(pre, A, hs);
    k3_pool <<<B_ * CH_,     256, 0, stream>>>(x, emb, hs, Wo, bo, Wg, bg, partial);
    k4_final<<<B_,           128, 0, stream>>>(partial, gamma, beta, out);
}